// GraphModule_16149077033381
// MI455X (gfx1250) — compile-verified
//
#include <hip/hip_runtime.h>

#define NN 100000
#define NE 1250000
#define IN_CH 128
#define HID 64
#define BN_EPS 1e-5f

typedef float v2f __attribute__((ext_vector_type(2)));
typedef float v8f __attribute__((ext_vector_type(8)));

// ---------------- degree / normalization ----------------

__global__ __launch_bounds__(256) void deg_init_kernel(float* deg, int n) {
  int i = blockIdx.x * blockDim.x + threadIdx.x;
  if (i < n) deg[i] = 1.0f;  // self loop contributes 1
}

__global__ __launch_bounds__(256) void deg_count_kernel(const int* __restrict__ dst,
                                                        float* __restrict__ deg, int e) {
  int i = blockIdx.x * blockDim.x + threadIdx.x;
  if (i < e) unsafeAtomicAdd(&deg[dst[i]], 1.0f);
}

__global__ __launch_bounds__(256) void dis_kernel(float* deg, int n) {
  int i = blockIdx.x * blockDim.x + threadIdx.x;
  if (i < n) deg[i] = rsqrtf(deg[i]);  // deg >= 1
}

// ---------------- WMMA GEMM: out[N,64] = A[N,K] @ B[K,64] (+fused BN/ReLU) ----------------
// One wave computes a 16-row x 64-col strip with V_WMMA_F32_16X16X4_F32.
// B is staged in LDS (padded stride to avoid bank conflicts).

template <int K, bool FUSE_BN>
__global__ __launch_bounds__(256) void gemm_wmma_kernel(
    const float* __restrict__ A, const float* __restrict__ B,
    const float* __restrict__ bias,
    const float* __restrict__ bng, const float* __restrict__ bnb,
    const float* __restrict__ bnm, const float* __restrict__ bnv,
    float* __restrict__ out, int n) {
  constexpr int LDB = HID + 1;  // pad to dodge LDS bank conflicts
  __shared__ float sB[K * LDB];

  for (int idx = threadIdx.x; idx < K * HID; idx += blockDim.x) {
    int r = idx / HID, c = idx - r * HID;
    sB[r * LDB + c] = B[idx];
  }
  __syncthreads();

  const int wave = threadIdx.x >> 5;
  const int lane = threadIdx.x & 31;
  const int half = lane >> 4;    // 0: lanes 0-15, 1: lanes 16-31
  const int lcol = lane & 15;
  const int rowTile = blockIdx.x * 8 + wave;
  if (rowTile * 16 >= n) return;

  v8f acc[4];
#pragma unroll
  for (int j = 0; j < 4; ++j)
#pragma unroll
    for (int g = 0; g < 8; ++g) acc[j][g] = 0.0f;

  // A fragment row for this lane (M = lane%16), K slice depends on half-wave.
  const float* arow = A + (size_t)(rowTile * 16 + lcol) * K;

  for (int k0 = 0; k0 < K; k0 += 4) {
    v2f a;
    a.x = arow[k0 + 2 * half];
    a.y = arow[k0 + 2 * half + 1];
#pragma unroll
    for (int j = 0; j < 4; ++j) {
      v2f b;
      b.x = sB[(k0 + 2 * half) * LDB + j * 16 + lcol];
      b.y = sB[(k0 + 2 * half + 1) * LDB + j * 16 + lcol];
      // D = A(16x4) * B(4x16) + C   (exact fp32 path)
      acc[j] = __builtin_amdgcn_wmma_f32_16x16x4_f32(
          false, a, false, b, (short)0, acc[j], false, false);
    }
  }

#pragma unroll
  for (int j = 0; j < 4; ++j) {
    const int c = j * 16 + lcol;
    float sc = 1.0f, sh = 0.0f, bi = 0.0f;
    if constexpr (FUSE_BN) {
      bi = bias[c];
      const float s = bng[c] * rsqrtf(bnv[c] + BN_EPS);
      sc = s;
      sh = bnb[c] - bnm[c] * s;
    }
#pragma unroll
    for (int g = 0; g < 8; ++g) {
      const int r = rowTile * 16 + g + 8 * half;
      float v = acc[j][g];
      if constexpr (FUSE_BN) {
        v = (v + bi) * sc + sh;
        v = fmaxf(v, 0.0f);
      }
      out[(size_t)r * HID + c] = v;
    }
  }
}

// ---------------- self-loop init: agg[n] = hw[n] * dis[n]^2 ----------------

__global__ __launch_bounds__(256) void agg_init_kernel(const float4* __restrict__ hw,
                                                       const float* __restrict__ dis,
                                                       float4* __restrict__ agg, int n) {
  int gid = blockIdx.x * blockDim.x + threadIdx.x;  // one float4 (4 ch) per thread
  int node = gid >> 4;
  if (node >= n) return;
  float w = dis[node];
  w = w * w;
  float4 v = hw[gid];
  agg[gid] = make_float4(v.x * w, v.y * w, v.z * w, v.w * w);
}

// ---------------- edge scatter: one wave per edge, 2 channels/lane ----------------

__global__ __launch_bounds__(256) void edge_scatter_kernel(
    const int* __restrict__ src, const int* __restrict__ dst,
    const float* __restrict__ dis, const float* __restrict__ hw,
    float* __restrict__ agg, int e) {
  int gid = blockIdx.x * blockDim.x + threadIdx.x;
  int edge = gid >> 5;
  int lane = gid & 31;
  if (edge >= e) return;
  const int s = src[edge];
  const int d = dst[edge];
  const float w = dis[s] * dis[d];
  const float2* hp = (const float2*)(hw + (size_t)s * HID);
  float2 v = hp[lane];
  float* ap = agg + (size_t)d * HID + lane * 2;
  unsafeAtomicAdd(ap, v.x * w);
  unsafeAtomicAdd(ap + 1, v.y * w);
}

// ---------------- post: out = relu(bn(agg + b)) + last ----------------

__global__ __launch_bounds__(256) void post_kernel(
    const float* __restrict__ agg, const float* __restrict__ bias,
    const float* __restrict__ bng, const float* __restrict__ bnb,
    const float* __restrict__ bnm, const float* __restrict__ bnv,
    const float* __restrict__ last, float* __restrict__ out, int n) {
  int gid = blockIdx.x * blockDim.x + threadIdx.x;  // 4 channels per thread
  int node = gid >> 4;
  if (node >= n) return;
  int base = (gid & 15) * 4;
  size_t off = (size_t)node * HID + base;
#pragma unroll
  for (int u = 0; u < 4; ++u) {
    int c = base + u;
    float s = bng[c] * rsqrtf(bnv[c] + BN_EPS);
    float v = (agg[off + u] + bias[c] - bnm[c]) * s + bnb[c];
    out[off + u] = fmaxf(v, 0.0f) + last[off + u];
  }
}

// ---------------- launch ----------------

extern "C" void kernel_launch(void* const* d_in, const int* in_sizes, int n_in,
                              void* d_out, int out_size, void* d_ws, size_t ws_size,
                              hipStream_t stream) {
  const float* x      = (const float*)d_in[0];
  const int*   ei     = (const int*)d_in[1];
  const int*   src    = ei;
  const int*   dst    = ei + NE;
  const float* fc_w   = (const float*)d_in[2];
  const float* fc_b   = (const float*)d_in[3];
  const float* conv_w = (const float*)d_in[4];  // [2,64,64]
  const float* conv_b = (const float*)d_in[5];  // [2,64]
  const float* bn_g   = (const float*)d_in[6];  // [3,64]
  const float* bn_b   = (const float*)d_in[7];
  const float* bn_m   = (const float*)d_in[8];
  const float* bn_v   = (const float*)d_in[9];
  float* out = (float*)d_out;

  float* dis  = (float*)d_ws;              // N floats (deg -> rsqrt(deg))
  float* last = dis + NN;                  // N*64  (post-FC activation, residual)
  float* hw   = last + (size_t)NN * HID;   // N*64  (h @ W)
  float* agg  = hw + (size_t)NN * HID;     // N*64  (segment sum)

  // 1) degree + normalization
  deg_init_kernel<<<(NN + 255) / 256, 256, 0, stream>>>(dis, NN);
  deg_count_kernel<<<(NE + 255) / 256, 256, 0, stream>>>(dst, dis, NE);
  dis_kernel<<<(NN + 255) / 256, 256, 0, stream>>>(dis, NN);

  // 2) FC + BN0 + ReLU -> last  (also the layer-1 input)
  const int rowTiles = (NN + 15) / 16;                 // 6250
  const int gemmBlocks = (rowTiles + 7) / 8;           // 8 waves/block
  gemm_wmma_kernel<IN_CH, true><<<gemmBlocks, 256, 0, stream>>>(
      x, fc_w, fc_b, bn_g, bn_b, bn_m, bn_v, last, NN);

  // 3) two GCN layers
  const float* hcur = last;
  for (int l = 0; l < 2; ++l) {
    gemm_wmma_kernel<HID, false><<<gemmBlocks, 256, 0, stream>>>(
        hcur, conv_w + (size_t)l * HID * HID, nullptr, nullptr, nullptr,
        nullptr, nullptr, hw, NN);
    agg_init_kernel<<<(NN * 16 + 255) / 256, 256, 0, stream>>>(
        (const float4*)hw, dis, (float4*)agg, NN);
    edge_scatter_kernel<<<((size_t)NE * 32 + 255) / 256, 256, 0, stream>>>(
        src, dst, dis, hw, agg, NE);
    post_kernel<<<(NN * 16 + 255) / 256, 256, 0, stream>>>(
        agg, conv_b + l * HID, bn_g + (l + 1) * HID, bn_b + (l + 1) * HID,
        bn_m + (l + 1) * HID, bn_v + (l + 1) * HID, last, out, NN);
    hcur = out;  // layer-2 input is layer-1 output (in d_out)
  }
}